// SPRGAT_88648124990305
// MI455X (gfx1250) — compile-verified
//
#include <hip/hip_runtime.h>
#include <hip/hip_bf16.h>

#define IN_DIM   128
#define HID      64
#define HEADS    4
#define FEAT     (HEADS * HID)   // 256
#define OUT_DIM  10
#define NGRAPHS  128

typedef __attribute__((ext_vector_type(2))) float v2f;
typedef __attribute__((ext_vector_type(8))) float v8f;

// ---------------------------------------------------------------------------
// Monotone float <-> uint key for atomicMax-based segment max.
// key(f) strictly increasing in f. key==0 is the minimum (acts as -inf init).
// ---------------------------------------------------------------------------
__device__ __forceinline__ unsigned fkey(float f) {
    unsigned b = __float_as_uint(f);
    return (b & 0x80000000u) ? ~b : (b | 0x80000000u);
}
__device__ __forceinline__ float funkey(unsigned k) {
    return (k & 0x80000000u) ? __uint_as_float(k & 0x7fffffffu)
                             : __uint_as_float(~k);
}

__device__ __forceinline__ v8f wmma4(v2f a, v2f b, v8f c) {
    return __builtin_amdgcn_wmma_f32_16x16x4_f32(
        /*neg_a=*/false, a, /*neg_b=*/false, b,
        /*c_mod=*/(short)0, c, /*reuse_a=*/false, /*reuse_b=*/false);
}

// ---------------------------------------------------------------------------
// fp32 WMMA GEMM: C[n_rows, 256] = A[n_rows, K] @ B[K, 256]   (row-major)
// One wave owns a 16x16 C tile; K is compile-time so B addressing folds into
// immediate offsets. Row index clamped (not predicated): OOB lanes read a
// valid row whose products land only in discarded C rows.
// Block = (32,4): 4 waves tile 64 consecutive columns; gridDim.y = 256/64.
// ---------------------------------------------------------------------------
template <int K>
__global__ __launch_bounds__(128)
void gat_gemm_wmma_f32(const float* __restrict__ A, const float* __restrict__ B,
                       float* __restrict__ C, int n_rows) {
    const int lane = threadIdx.x;                       // 0..31
    const int wave = threadIdx.y;                       // 0..3
    const int row0 = blockIdx.x * 16;
    const int col0 = (blockIdx.y * 4 + wave) * 16;

    const int m  = lane & 15;          // A row / B,C col within tile
    const int kh = lane >> 4;          // K half-select (0 -> k{0,1}, 1 -> k{2,3})

    int r = row0 + m;
    if (r >= n_rows) r = n_rows - 1;   // clamp: OOB rows never stored

    const float* Ap = A + (size_t)r * K + 2 * kh;            // +=16 per chunk
    const float* Bp = B + (size_t)(2 * kh) * FEAT + col0 + m; // +=16*FEAT per chunk

    v8f acc = {};
    #pragma unroll
    for (int k = 0; k < K; k += 16) {
        v2f a0 = *(const v2f*)(Ap + 0);
        v2f a1 = *(const v2f*)(Ap + 4);
        v2f a2 = *(const v2f*)(Ap + 8);
        v2f a3 = *(const v2f*)(Ap + 12);
        v2f b0; b0.x = Bp[ 0 * FEAT]; b0.y = Bp[ 1 * FEAT];
        v2f b1; b1.x = Bp[ 4 * FEAT]; b1.y = Bp[ 5 * FEAT];
        v2f b2; b2.x = Bp[ 8 * FEAT]; b2.y = Bp[ 9 * FEAT];
        v2f b3; b3.x = Bp[12 * FEAT]; b3.y = Bp[13 * FEAT];
        acc = wmma4(a0, b0, acc);
        acc = wmma4(a1, b1, acc);
        acc = wmma4(a2, b2, acc);
        acc = wmma4(a3, b3, acc);
        Ap += 16;
        Bp += 16 * FEAT;
    }

    const int nn = lane & 15;
    const int mb = (lane >> 4) * 8;    // C: lanes 16-31 hold M = i+8
    #pragma unroll
    for (int i = 0; i < 8; ++i) {
        const int rr = row0 + mb + i;
        if (rr < n_rows) C[(size_t)rr * FEAT + col0 + nn] = acc[i];
    }
}

// ---------------------------------------------------------------------------
// alpha_s[n,h] = sum_c h[n,h,c]*a_src[h,c];  alpha_d likewise. 1 thread/(n,h).
// ---------------------------------------------------------------------------
__global__ void gat_alpha(const float* __restrict__ h,
                          const float* __restrict__ a_src,
                          const float* __restrict__ a_dst,
                          float* __restrict__ as_, float* __restrict__ ad_, int n) {
    int i = blockIdx.x * blockDim.x + threadIdx.x;
    if (i >= n * HEADS) return;
    int node = i >> 2, hd = i & 3;
    const float* hp = h + (size_t)node * FEAT + hd * HID;
    const float* sp = a_src + hd * HID;
    const float* dp = a_dst + hd * HID;
    float s = 0.f, d = 0.f;
    #pragma unroll 8
    for (int c = 0; c < HID; ++c) { float v = hp[c]; s += v * sp[c]; d += v * dp[c]; }
    as_[i] = s; ad_[i] = d;
}

// ---------------------------------------------------------------------------
// Pass 1: segment max of leaky-relu scores per (dst, head) via keyed atomicMax.
// Edges e<E come from edge_index; e>=E are self loops (node e-E).
// ---------------------------------------------------------------------------
__global__ void gat_edge_max(const int* __restrict__ src, const int* __restrict__ dst,
                             int E, int n,
                             const float* __restrict__ as_, const float* __restrict__ ad_,
                             unsigned* __restrict__ mkey) {
    int i = blockIdx.x * blockDim.x + threadIdx.x;
    int total = (E + n) * HEADS;
    if (i >= total) return;
    int e = i >> 2, hd = i & 3;
    int s = (e < E) ? src[e] : (e - E);
    int d = (e < E) ? dst[e] : (e - E);
    float sc = as_[s * HEADS + hd] + ad_[d * HEADS + hd];
    sc = (sc > 0.f) ? sc : sc * 0.2f;
    atomicMax(&mkey[d * HEADS + hd], fkey(sc));
}

// ---------------------------------------------------------------------------
// Pass 2: e = exp(score - max); num[dst,h,c] += e*h[src,h,c]; den[dst,h] += e.
// One 256-thread block per edge: t = h*64 + c.
// ---------------------------------------------------------------------------
__global__ __launch_bounds__(FEAT)
void gat_edge_agg(const int* __restrict__ src, const int* __restrict__ dst,
                  int E, int n,
                  const float* __restrict__ as_, const float* __restrict__ ad_,
                  const unsigned* __restrict__ mkey,
                  const float* __restrict__ h,
                  float* __restrict__ num, float* __restrict__ den) {
    int e = blockIdx.x;
    int t = threadIdx.x;
    int hd = t >> 6;
    int s = (e < E) ? src[e] : (e - E);
    int d = (e < E) ? dst[e] : (e - E);
    float sc = as_[s * HEADS + hd] + ad_[d * HEADS + hd];
    sc = (sc > 0.f) ? sc : sc * 0.2f;
    float mval = funkey(mkey[d * HEADS + hd]);
    float ev = __expf(sc - mval);
    atomicAdd(&num[(size_t)d * FEAT + t], ev * h[(size_t)s * FEAT + t]);
    if ((t & 63) == 0) atomicAdd(&den[d * HEADS + hd], ev);
}

// Layer-1 finalize: out = relu(num/(den+eps) + b1)   (concat heads -> 256)
__global__ void gat_norm1(const float* __restrict__ num, const float* __restrict__ den,
                          const float* __restrict__ b1, float* __restrict__ out, int n) {
    int i = blockIdx.x * blockDim.x + threadIdx.x;
    if (i >= n * FEAT) return;
    int node = i >> 8, t = i & 255, hd = t >> 6;
    float v = num[i] / (den[node * HEADS + hd] + 1e-16f) + b1[t];
    out[i] = v > 0.f ? v : 0.f;
}

// Layer-2 finalize: mean over heads -> 64ch, + b2, relu
__global__ void gat_norm2(const float* __restrict__ num, const float* __restrict__ den,
                          const float* __restrict__ b2, float* __restrict__ out, int n) {
    int i = blockIdx.x * blockDim.x + threadIdx.x;
    if (i >= n * HID) return;
    int node = i >> 6, c = i & 63;
    float acc = 0.f;
    #pragma unroll
    for (int hd = 0; hd < HEADS; ++hd)
        acc += num[(size_t)node * FEAT + hd * HID + c] / (den[node * HEADS + hd] + 1e-16f);
    float v = acc * 0.25f + b2[c];
    out[i] = v > 0.f ? v : 0.f;
}

// Global mean pool (accumulate)
__global__ void gat_pool(const float* __restrict__ h2, const int* __restrict__ batch,
                         float* __restrict__ pool, float* __restrict__ cnt, int n) {
    int i = blockIdx.x * blockDim.x + threadIdx.x;
    if (i >= n * HID) return;
    int node = i >> 6, c = i & 63;
    int g = batch[node];
    atomicAdd(&pool[g * HID + c], h2[i]);
    if (c == 0) atomicAdd(&cnt[g], 1.0f);
}

// Final linear head: out[g,o] = (pool[g,:]/max(cnt,1)) . lin_w[:,o] + lin_b[o]
__global__ void gat_head(const float* __restrict__ pool, const float* __restrict__ cnt,
                         const float* __restrict__ lw, const float* __restrict__ lb,
                         float* __restrict__ out) {
    int i = blockIdx.x * blockDim.x + threadIdx.x;
    if (i >= NGRAPHS * OUT_DIM) return;
    int g = i / OUT_DIM, o = i % OUT_DIM;
    float c = cnt[g]; c = c > 1.0f ? c : 1.0f;
    float acc = 0.f;
    #pragma unroll 8
    for (int k = 0; k < HID; ++k) acc += (pool[g * HID + k] / c) * lw[k * OUT_DIM + o];
    out[i] = acc + lb[o];
}

// ---------------------------------------------------------------------------
extern "C" void kernel_launch(void* const* d_in, const int* in_sizes, int n_in,
                              void* d_out, int out_size, void* d_ws, size_t ws_size,
                              hipStream_t stream) {
    (void)n_in; (void)out_size; (void)ws_size;

    const float* x      = (const float*)d_in[0];
    const int*   eidx   = (const int*)  d_in[1];
    const int*   batch  = (const int*)  d_in[2];
    const float* W1     = (const float*)d_in[3];
    const float* a_src1 = (const float*)d_in[4];
    const float* a_dst1 = (const float*)d_in[5];
    const float* b1     = (const float*)d_in[6];
    const float* W2     = (const float*)d_in[7];
    const float* a_src2 = (const float*)d_in[8];
    const float* a_dst2 = (const float*)d_in[9];
    const float* b2     = (const float*)d_in[10];
    const float* lin_w  = (const float*)d_in[11];
    const float* lin_b  = (const float*)d_in[12];
    float* out = (float*)d_out;

    const int N = in_sizes[0] / IN_DIM;       // 25000
    const int E = in_sizes[1] / 2;            // 400000
    const int ET = E + N;                     // edges incl. self loops
    const int* src = eidx;
    const int* dst = eidx + E;

    // Workspace layout (floats)
    float* ws = (float*)d_ws;
    size_t off = 0;
    float*    hpre = ws + off; off += (size_t)N * FEAT;   // GEMM output (reused layer 2)
    float*    num  = ws + off; off += (size_t)N * FEAT;   // softmax numerator accum
    float*    out1 = ws + off; off += (size_t)N * FEAT;   // layer-1 output
    float*    h2   = ws + off; off += (size_t)N * HID;    // layer-2 output
    float*    as_  = ws + off; off += (size_t)N * HEADS;
    float*    ad_  = ws + off; off += (size_t)N * HEADS;
    float*    den  = ws + off; off += (size_t)N * HEADS;
    unsigned* mkey = (unsigned*)(ws + off); off += (size_t)N * HEADS;
    float*    pool = ws + off; off += (size_t)NGRAPHS * HID;
    float*    cnt  = ws + off; off += (size_t)NGRAPHS;

    const int rowTiles = (N + 15) / 16;
    dim3 gemmBlock(32, 4);
    dim3 gemmGrid(rowTiles, FEAT / 64);       // 4 waves/block * 16 cols = 64 cols

    const int ethreads = ET * HEADS;

    // ----- Layer 1 -----
    gat_gemm_wmma_f32<IN_DIM><<<gemmGrid, gemmBlock, 0, stream>>>(x, W1, hpre, N);
    gat_alpha<<<(N * HEADS + 255) / 256, 256, 0, stream>>>(hpre, a_src1, a_dst1, as_, ad_, N);
    hipMemsetAsync(mkey, 0, (size_t)N * HEADS * sizeof(unsigned), stream);
    hipMemsetAsync(den,  0, (size_t)N * HEADS * sizeof(float),    stream);
    hipMemsetAsync(num,  0, (size_t)N * FEAT  * sizeof(float),    stream);
    gat_edge_max<<<(ethreads + 255) / 256, 256, 0, stream>>>(src, dst, E, N, as_, ad_, mkey);
    gat_edge_agg<<<ET, FEAT, 0, stream>>>(src, dst, E, N, as_, ad_, mkey, hpre, num, den);
    gat_norm1<<<((size_t)N * FEAT + 255) / 256, 256, 0, stream>>>(num, den, b1, out1, N);

    // ----- Layer 2 -----
    gat_gemm_wmma_f32<FEAT><<<gemmGrid, gemmBlock, 0, stream>>>(out1, W2, hpre, N);
    gat_alpha<<<(N * HEADS + 255) / 256, 256, 0, stream>>>(hpre, a_src2, a_dst2, as_, ad_, N);
    hipMemsetAsync(mkey, 0, (size_t)N * HEADS * sizeof(unsigned), stream);
    hipMemsetAsync(den,  0, (size_t)N * HEADS * sizeof(float),    stream);
    hipMemsetAsync(num,  0, (size_t)N * FEAT  * sizeof(float),    stream);
    gat_edge_max<<<(ethreads + 255) / 256, 256, 0, stream>>>(src, dst, E, N, as_, ad_, mkey);
    gat_edge_agg<<<ET, FEAT, 0, stream>>>(src, dst, E, N, as_, ad_, mkey, hpre, num, den);
    gat_norm2<<<((size_t)N * HID + 255) / 256, 256, 0, stream>>>(num, den, b2, h2, N);

    // ----- Pool + head -----
    hipMemsetAsync(pool, 0, (size_t)NGRAPHS * HID * sizeof(float), stream);
    hipMemsetAsync(cnt,  0, (size_t)NGRAPHS * sizeof(float),       stream);
    gat_pool<<<((size_t)N * HID + 255) / 256, 256, 0, stream>>>(h2, batch, pool, cnt, N);
    gat_head<<<(NGRAPHS * OUT_DIM + 255) / 256, 256, 0, stream>>>(pool, cnt, lin_w, lin_b, out);
}